// MultiboxDetector_70669391888766
// MI455X (gfx1250) — compile-verified
//
#include <hip/hip_runtime.h>

typedef float v2f __attribute__((ext_vector_type(2)));
typedef float v8f __attribute__((ext_vector_type(8)));

#define NCLS 81        // NC+1
#define TOT_ANCH 8732
#define BATCH 8
#define CB 64          // ci block staged in LDS per step (divides 256/512/1024)
#define PITCH 160      // LDS row pitch (words) for ci-pair rows: conflict-free b64
#define TILEWORDS ((CB / 2) * PITCH)
#define TILEBYTES (TILEWORDS * 4)

struct ConvArgs {
    const float* __restrict__ x;     // (B, C, H, W)
    const float* __restrict__ w;     // (Cout, C, 3, 3)
    const float* __restrict__ bias;  // (Cout,)
    float* __restrict__ out;         // base of loc or conf region of d_out
    int C, H, W;
    int Cout;
    int nb;
    int L;          // last-dim stride: 4 (loc) or 81 (conf)
    int anchorOff;  // cumulative nb*H*W offset of this scale
    int HW;
    int Ncols;      // B*H*W
    int K;          // C*9
    int cbl;        // log2(C / CB)
};

__device__ __forceinline__ void async_load_lds_b32(unsigned lds_byte_addr,
                                                   const float* gptr) {
    // CDNA5 async DMA: global -> LDS, one b32 per lane, tracked by ASYNCcnt.
    asm volatile("global_load_async_to_lds_b32 %0, %1, off"
                 :
                 : "v"(lds_byte_addr), "v"(gptr)
                 : "memory");
}

__device__ __forceinline__ void wait_asynccnt0() {
    asm volatile("s_wait_asynccnt 0x0" ::: "memory");
}

// Implicit-GEMM 3x3 SAME conv via V_WMMA_F32_16X16X4_F32.
//   M = Cout (A = weights), N = B*H*W (B = im2col), K = C*9 ordered (kh,kw)-major.
// Block = 256 threads (8 waves); all waves share one 64-column B tile in LDS,
// each wave owns one 16-row M tile. Ping-pong double buffer: async global->LDS
// DMA for tile t+1 overlaps the WMMA sweep over tile t; one barrier per tile.
__global__ __launch_bounds__(256) void conv3x3_wmma_f32(ConvArgs a) {
    __shared__ float tile[2][TILEWORDS];   // 2 x 20 KB

    const int tid = threadIdx.x;
    const int lane = tid & 31;
    const int wave = tid >> 5;
    const int lmod = lane & 15;
    const int lhalf = lane >> 4;      // 0 or 1 (K sub-pair select)

    const int mtiles = (a.Cout + 15) >> 4;
    int tileM = blockIdx.y * 8 + wave;
    if (tileM >= mtiles) tileM = mtiles - 1;   // clamp; stores are guarded
    const int row = tileM * 16 + lmod;
    const int rowc = (row < a.Cout) ? row : 0;

    // ---- staging coordinates: 256 threads cover 64 cols x 16 ci each ----
    const int scol = tid & 63;
    const int scia = (tid >> 6) * (CB / 4);    // 0,16,32,48

    int sN = blockIdx.x * 64 + scol;
    const bool sColV = sN < a.Ncols;
    int snc = sColV ? sN : 0;
    const int sB = snc / a.HW;
    int ssp = snc - sB * a.HW;
    const int sOh = ssp / a.W;
    const int sOw = ssp - sOh * a.W;

    const unsigned tileBase = (unsigned)(uintptr_t)(void*)&tile[0][0];

    // ---- compute-column decode (needed only for the scatter stores) ----
    int n0, n1, n2, n3, b0, b1, b2, b3, sp0, sp1, sp2, sp3;
    bool v0, v1, v2, v3;
#define DECODE_COL(s, nn, bb, spv, vv)              \
    {                                               \
        nn = blockIdx.x * 64 + (s)*16 + lmod;       \
        vv = nn < a.Ncols;                          \
        int nc = vv ? nn : 0;                       \
        bb = nc / a.HW;                             \
        spv = nc - bb * a.HW;                       \
    }
    DECODE_COL(0, n0, b0, sp0, v0)
    DECODE_COL(1, n1, b1, sp1, v1)
    DECODE_COL(2, n2, b2, sp2, v2)
    DECODE_COL(3, n3, b3, sp3, v3)
#undef DECODE_COL

    const int HW = a.HW;
    const int nblkMask = (1 << a.cbl) - 1;
    const int T = 9 << a.cbl;   // total K tiles

    // Stage K-tile t into buffer `buf` (issue only; caller waits/syncs).
    auto stageTile = [&](int t, int buf) {
        const int r = t >> a.cbl;
        const int cb = (t & nblkMask) * CB;
        const int dh = r / 3 - 1;
        const int dw = r - (r / 3) * 3 - 1;
        const int ih = sOh + dh;
        const int iw = sOw + dw;
        const bool sval = sColV && ((unsigned)ih < (unsigned)a.H) &&
                          ((unsigned)iw < (unsigned)a.W);
        if (sval) {
            const float* __restrict__ sb =
                a.x + (ptrdiff_t)((sB * a.C + cb) * a.H + ih) * a.W + iw;
            const unsigned lbase = tileBase + (unsigned)buf * TILEBYTES;
#pragma unroll
            for (int i = 0; i < CB / 4; ++i) {
                const int cil = scia + i;
                const unsigned widx =
                    (unsigned)((cil >> 1) * PITCH + scol * 2 + (cil & 1));
                async_load_lds_b32(lbase + widx * 4u,
                                   sb + (ptrdiff_t)cil * HW);
            }
        } else {
            float* __restrict__ tp = &tile[buf][0];
#pragma unroll
            for (int i = 0; i < CB / 4; ++i) {
                const int cil = scia + i;
                tp[(cil >> 1) * PITCH + scol * 2 + (cil & 1)] = 0.0f;
            }
        }
    };

    v8f acc0 = {}, acc1 = {}, acc2 = {}, acc3 = {};

    stageTile(0, 0);
    wait_asynccnt0();
    __syncthreads();

    for (int t = 0; t < T; ++t) {
        const int buf = t & 1;
        if (t + 1 < T) stageTile(t + 1, buf ^ 1);   // overlap DMA with compute

        const int r = t >> a.cbl;
        const int cb = (t & nblkMask) * CB;
        // per-lane weight base: constant k*9 indices below fold into load offsets
        const float* __restrict__ wp =
            a.w + (size_t)rowc * a.K + r + (size_t)(cb + lhalf * 2) * 9;
        const float* __restrict__ tb = &tile[buf][0];

#pragma unroll
        for (int k = 0; k < CB; k += 4) {
            v2f av;
            av.x = wp[k * 9];
            av.y = wp[k * 9 + 9];
            const float* __restrict__ trow = tb + ((k >> 1) + lhalf) * PITCH;

            v2f bv;
            bv = *(const v2f*)(trow + lmod * 2);
            acc0 = __builtin_amdgcn_wmma_f32_16x16x4_f32(
                false, av, false, bv, (short)0, acc0, false, false);

            bv = *(const v2f*)(trow + (lmod + 16) * 2);
            acc1 = __builtin_amdgcn_wmma_f32_16x16x4_f32(
                false, av, false, bv, (short)0, acc1, false, false);

            bv = *(const v2f*)(trow + (lmod + 32) * 2);
            acc2 = __builtin_amdgcn_wmma_f32_16x16x4_f32(
                false, av, false, bv, (short)0, acc2, false, false);

            bv = *(const v2f*)(trow + (lmod + 48) * 2);
            acc3 = __builtin_amdgcn_wmma_f32_16x16x4_f32(
                false, av, false, bv, (short)0, acc3, false, false);
        }

        wait_asynccnt0();   // next tile fully in LDS (this wave's DMAs)
        __syncthreads();    // everyone done staging + done reading tile[buf]
    }

    // ---- epilogue: bias + scatter into (B, 8732, L) layout ----
    // C/D layout: VGPR e, lanes 0-15 -> M=e; lanes 16-31 -> M=e+8; N = lane&15.
    const int mbase = tileM * 16 + lhalf * 8;
#define STORE_SUB(accv, nn, bb, spv, vv)                                            \
    if (vv) {                                                                       \
        size_t base = (size_t)bb * ((size_t)TOT_ANCH * a.L);                        \
        _Pragma("unroll") for (int e = 0; e < 8; ++e) {                             \
            int co = mbase + e;                                                     \
            if (co < a.Cout) {                                                      \
                int cp = co % a.nb;                                                 \
                int j = co / a.nb;                                                  \
                size_t idx = base + (size_t)(a.anchorOff + cp * a.HW + spv) * a.L + j; \
                a.out[idx] = accv[e] + a.bias[co];                                  \
            }                                                                       \
        }                                                                           \
    }
    STORE_SUB(acc0, n0, b0, sp0, v0)
    STORE_SUB(acc1, n1, b1, sp1, v1)
    STORE_SUB(acc2, n2, b2, sp2, v2)
    STORE_SUB(acc3, n3, b3, sp3, v3)
#undef STORE_SUB
}

extern "C" void kernel_launch(void* const* d_in, const int* in_sizes, int n_in,
                              void* d_out, int out_size, void* d_ws, size_t ws_size,
                              hipStream_t stream) {
    static const int CHANS[6] = {512, 1024, 512, 256, 256, 256};
    static const int RESH[6] = {38, 19, 10, 5, 3, 1};
    static const int NB[6] = {4, 6, 6, 6, 4, 4};
    static const int CBL[6] = {3, 4, 3, 2, 2, 2};   // log2(C/64)

    float* out = (float*)d_out;
    float* locOut = out;
    float* confOut = out + (size_t)BATCH * TOT_ANCH * 4;

    int aoff = 0;
    for (int i = 0; i < 6; ++i) {
        const float* x = (const float*)d_in[5 * i + 0];
        const float* lw = (const float*)d_in[5 * i + 1];
        const float* lb = (const float*)d_in[5 * i + 2];
        const float* cw = (const float*)d_in[5 * i + 3];
        const float* cb = (const float*)d_in[5 * i + 4];

        const int C = CHANS[i], H = RESH[i], W = RESH[i], nb = NB[i];
        const int HW = H * W;
        const int Ncols = BATCH * HW;
        const int K = C * 9;

        // loc head
        {
            ConvArgs a{x, lw, lb, locOut, C, H, W, nb * 4, nb, 4,
                       aoff, HW, Ncols, K, CBL[i]};
            int mtiles = (nb * 4 + 15) / 16;
            dim3 grid((Ncols + 63) / 64, (mtiles + 7) / 8);
            conv3x3_wmma_f32<<<grid, 256, 0, stream>>>(a);
        }
        // conf head
        {
            ConvArgs a{x, cw, cb, confOut, C, H, W, nb * NCLS, nb, NCLS,
                       aoff, HW, Ncols, K, CBL[i]};
            int mtiles = (nb * NCLS + 15) / 16;
            dim3 grid((Ncols + 63) / 64, (mtiles + 7) / 8);
            conv3x3_wmma_f32<<<grid, 256, 0, stream>>>(a);
        }

        aoff += nb * HW;
    }
}